// Classifier_1151051235754
// MI455X (gfx1250) — compile-verified
//
#include <hip/hip_runtime.h>
#include <math.h>

typedef __attribute__((ext_vector_type(2))) float v2f;
typedef __attribute__((ext_vector_type(8))) float v8f;

#define GEMM_TPB 4   // row super-tiles (32 rows each) per block

// ---------------------------------------------------------------------------
// Z = A(NxK) @ B(KxH), fp32 WMMA 16x16x4, LDS-staged operands.
// Block: 256 threads = 8 waves = 2 row-tiles x 4 col-tiles (32x64 super-tile).
// Grid: ( ceil(N / (32*GEMM_TPB)), Hn/64 ).
// LDS: Bt[64][K+4] (W transposed) + As[32][K+4], stride padded so the v2f
// fragment loads are bank-conflict-free ds_load_b64.
// ---------------------------------------------------------------------------
__global__ __launch_bounds__(256) void gemm_wmma_f32(
    const float* __restrict__ A, const float* __restrict__ B,
    float* __restrict__ Z, int N, int K, int Hn)
{
    extern __shared__ float lds[];
    const int KP = K + 4;
    float* lsBt = lds;                       // 64 * KP
    float* lsA  = lds + (size_t)64 * KP;     // 32 * KP

    const int tid  = threadIdx.x;
    const int wave = tid >> 5;
    const int lane = tid & 31;
    const int g    = lane >> 4;              // lane group 0/1
    const int ml   = lane & 15;
    const int rowT = wave >> 2;              // 0/1 : which 16-row tile
    const int colT = wave & 3;               // 0..3: which 16-col tile
    const int hBase = blockIdx.y * 64;       // global column base of this block

    // Stage W columns [hBase, hBase+64) transposed: lsBt[hh*KP + k] = B[k][hBase+hh]
    for (int idx = tid; idx < K * 64; idx += 256) {
        int k  = idx >> 6;                   // idx / 64
        int hh = idx & 63;                   // idx % 64
        lsBt[hh * KP + k] = B[(size_t)k * Hn + hBase + hh];
    }

    const float* ap0 = lsA  + (rowT * 16 + ml) * KP + 2 * g;
    const float* bp0 = lsBt + (colT * 16 + ml) * KP + 2 * g;

    for (int rt = 0; rt < GEMM_TPB; ++rt) {
        int r0 = (blockIdx.x * GEMM_TPB + rt) * 32;     // 32-row super-tile base
        if (r0 >= N) break;                              // uniform
        __syncthreads();                                 // Bt ready / prev tile done
        // Stage A rows [r0, r0+32), zero-filled past N
        for (int idx = tid; idx < 32 * K; idx += 256) {
            int r = idx / K, k = idx - r * K;
            int row = r0 + r;
            lsA[r * KP + k] = (row < N) ? A[(size_t)row * K + k] : 0.0f;
        }
        __syncthreads();

        v8f acc = {};
#pragma unroll 8
        for (int k = 0; k < K; k += 4) {
            v2f a = *(const v2f*)(ap0 + k);
            v2f b = *(const v2f*)(bp0 + k);
            acc = __builtin_amdgcn_wmma_f32_16x16x4_f32(
                false, a, false, b, (short)0, acc, false, false);
        }

        int rBase = r0 + rowT * 16;
        int cIdx  = hBase + colT * 16 + ml;
#pragma unroll
        for (int v = 0; v < 8; ++v) {                    // D: row = v + 8g
            int row = rBase + v + 8 * g;
            if (row < N) Z[(size_t)row * Hn + cIdx] = acc[v];
        }
    }
}

// ---------------------------------------------------------------------------
// Per-node attention scores: s_src[i] = z[i].a[0:H], s_dst[i] = z[i].a[H:2H]
// One wave per node.
// ---------------------------------------------------------------------------
__global__ __launch_bounds__(256) void node_scores(
    const float* __restrict__ z, const float* __restrict__ a,
    float* __restrict__ ssrc, float* __restrict__ sdst, int N, int Hn)
{
    int wid  = blockIdx.x * (blockDim.x >> 5) + (threadIdx.x >> 5);
    int lane = threadIdx.x & 31;
    if (wid >= N) return;
    const float* zr = z + (size_t)wid * Hn;
    float p0 = 0.f, p1 = 0.f;
    for (int j = lane; j < Hn; j += 32) {
        float v = zr[j];
        p0 += v * a[j];
        p1 += v * a[Hn + j];
    }
    for (int off = 16; off; off >>= 1) {
        p0 += __shfl_xor(p0, off, 32);
        p1 += __shfl_xor(p1, off, 32);
    }
    if (lane == 0) { ssrc[wid] = p0; sdst[wid] = p1; }
}

// ---------------------------------------------------------------------------
// Per-layer reset: zero h_next (N*H), hg slab (G*H), cnt (G); nmax=-inf, nsum=0.
// ---------------------------------------------------------------------------
__global__ void reset_layer(float* __restrict__ hnext, float* __restrict__ nmax,
                            float* __restrict__ nsum, float* __restrict__ hg,
                            float* __restrict__ cnt, int N, int Hn, int G)
{
    long long idx = (long long)blockIdx.x * blockDim.x + threadIdx.x;
    long long tot = (long long)N * Hn;
    if (idx < tot) hnext[idx] = 0.0f;
    if (idx < N) { nmax[idx] = __int_as_float(0xff800000); nsum[idx] = 0.0f; }
    if (idx < (long long)G * Hn) hg[idx] = 0.0f;
    if (idx < G) cnt[idx] = 0.0f;
}

__device__ __forceinline__ void atomicMaxFloat(float* addr, float val) {
    if (val >= 0.0f) atomicMax((int*)addr, __float_as_int(val));
    else             atomicMin((unsigned int*)addr, __float_as_uint(val));
}

// ---------------------------------------------------------------------------
// Edge logits + segment max over dst. One thread per edge.
// ---------------------------------------------------------------------------
__global__ void edge_logits(const float* __restrict__ ef, const float* __restrict__ ae,
                            const int* __restrict__ src, const int* __restrict__ dst,
                            const float* __restrict__ ssrc, const float* __restrict__ sdst,
                            float* __restrict__ elog, float* __restrict__ nmax,
                            int E, int ED)
{
    int e = blockIdx.x * blockDim.x + threadIdx.x;
    if (e >= E) return;
    int d = dst[e];
    float t = ssrc[src[e]] + sdst[d];
    const float* er = ef + (size_t)e * ED;
    for (int j = 0; j < ED; ++j) t += er[j] * ae[j];
    t = (t >= 0.0f) ? t : 0.2f * t;            // leaky_relu(0.2)
    elog[e] = t;
    atomicMaxFloat(&nmax[d], t);
}

// ---------------------------------------------------------------------------
// ex = exp(logit - max[dst]); segment sum. One thread per edge.
// ---------------------------------------------------------------------------
__global__ void edge_exp(const int* __restrict__ dst, float* __restrict__ elog,
                         const float* __restrict__ nmax, float* __restrict__ nsum, int E)
{
    int e = blockIdx.x * blockDim.x + threadIdx.x;
    if (e >= E) return;
    int d = dst[e];
    float m = nmax[d];
    if (m < -3.0e38f) m = 0.0f;                // isfinite clamp (-inf case)
    float ex = expf(elog[e] - m);
    elog[e] = ex;
    atomicAdd(&nsum[d], ex);
}

// ---------------------------------------------------------------------------
// Message aggregation: h_next[dst] += alpha * z[src]. One wave per edge,
// float4 per lane (128 floats / 32 lanes).
// ---------------------------------------------------------------------------
__global__ __launch_bounds__(256) void aggregate(
    const float* __restrict__ z, const int* __restrict__ src,
    const int* __restrict__ dst, const float* __restrict__ elog,
    const float* __restrict__ nsum, float* __restrict__ hout, int E, int Hn)
{
    int wid  = blockIdx.x * (blockDim.x >> 5) + (threadIdx.x >> 5);
    int lane = threadIdx.x & 31;
    if (wid >= E) return;
    int s = src[wid], d = dst[wid];
    float den   = nsum[d];
    float alpha = elog[wid] / (den > 0.0f ? den : 1.0f);
    const float* zs = z + (size_t)s * Hn;
    float*       hd = hout + (size_t)d * Hn;
    for (int j = lane * 4; j < Hn; j += 128) {
        float4 v = *(const float4*)(zs + j);
        atomicAdd(hd + j + 0, alpha * v.x);
        atomicAdd(hd + j + 1, alpha * v.y);
        atomicAdd(hd + j + 2, alpha * v.z);
        atomicAdd(hd + j + 3, alpha * v.w);
    }
}

__global__ void relu_inplace(float* __restrict__ h, long long n)
{
    long long idx = (long long)blockIdx.x * blockDim.x + threadIdx.x;
    if (idx < n) h[idx] = fmaxf(h[idx], 0.0f);
}

// ---------------------------------------------------------------------------
// Readout accumulate: att = exp(leaky(h.w_att + b, 0.01));
// num[g] += att*h; cnt[g] += 1. One wave per node.
// ---------------------------------------------------------------------------
__global__ __launch_bounds__(256) void readout_accum(
    const float* __restrict__ h, const int* __restrict__ gid,
    const float* __restrict__ watt, const float* __restrict__ batt,
    float* __restrict__ num, float* __restrict__ cnt, int N, int Hn)
{
    int wid  = blockIdx.x * (blockDim.x >> 5) + (threadIdx.x >> 5);
    int lane = threadIdx.x & 31;
    if (wid >= N) return;
    const float* hr = h + (size_t)wid * Hn;
    float p = 0.f;
    for (int j = lane; j < Hn; j += 32) p += hr[j] * watt[j];
    for (int off = 16; off; off >>= 1) p += __shfl_xor(p, off, 32);
    float t = p + batt[0];
    t = (t >= 0.0f) ? t : 0.01f * t;
    float att = expf(t);
    int g = gid[wid];
    float* ng = num + (size_t)g * Hn;
    for (int j = lane * 4; j < Hn; j += 128) {
        float4 v = *(const float4*)(hr + j);
        atomicAdd(ng + j + 0, att * v.x);
        atomicAdd(ng + j + 1, att * v.y);
        atomicAdd(ng + j + 2, att * v.z);
        atomicAdd(ng + j + 3, att * v.w);
    }
    if (lane == 0) atomicAdd(&cnt[g], 1.0f);
}

__global__ void readout_finalize(float* __restrict__ hg, const float* __restrict__ cnt,
                                 int G, int Hn)
{
    int idx = blockIdx.x * blockDim.x + threadIdx.x;
    if (idx < G * Hn) hg[idx] /= fmaxf(cnt[idx / Hn], 1.0f);
}

// ---------------------------------------------------------------------------
// Classifier: y = concat(hg1,hg2,hg3) @ W_cls + b_cls; log_softmax.
// One wave per graph. hg layout: [3][G][Hn].
// ---------------------------------------------------------------------------
__global__ __launch_bounds__(32) void classifier(
    const float* __restrict__ hg, const float* __restrict__ Wc,
    const float* __restrict__ bc, float* __restrict__ out, int G, int Hn, int C)
{
    int g = blockIdx.x;
    int lane = threadIdx.x;
    float acc[16];
    for (int c = 0; c < C; ++c) acc[c] = 0.0f;
    int F = 3 * Hn;
    for (int k = lane; k < F; k += 32) {
        int slab = k / Hn, w = k % Hn;
        float f = hg[(size_t)slab * G * Hn + (size_t)g * Hn + w];
        const float* wr = Wc + (size_t)k * C;
        for (int c = 0; c < C; ++c) acc[c] += f * wr[c];
    }
    for (int off = 16; off; off >>= 1)
        for (int c = 0; c < C; ++c) acc[c] += __shfl_xor(acc[c], off, 32);
    if (lane == 0) {
        float m = -3.0e38f;
        for (int c = 0; c < C; ++c) { acc[c] += bc[c]; m = fmaxf(m, acc[c]); }
        float s = 0.0f;
        for (int c = 0; c < C; ++c) s += expf(acc[c] - m);
        float lse = m + logf(s);
        for (int c = 0; c < C; ++c) out[(size_t)g * C + c] = acc[c] - lse;
    }
}

// ---------------------------------------------------------------------------
extern "C" void kernel_launch(void* const* d_in, const int* in_sizes, int n_in,
                              void* d_out, int out_size, void* d_ws, size_t ws_size,
                              hipStream_t stream)
{
    const float* x    = (const float*)d_in[0];
    const float* ef   = (const float*)d_in[1];
    const int*   src  = (const int*)d_in[2];
    const int*   dst  = (const int*)d_in[3];
    const int*   gid  = (const int*)d_in[4];
    const float* W1   = (const float*)d_in[5];
    const float* a1   = (const float*)d_in[6];
    const float* W2   = (const float*)d_in[7];
    const float* a2   = (const float*)d_in[8];
    const float* W3   = (const float*)d_in[9];
    const float* a3   = (const float*)d_in[10];
    const float* watt = (const float*)d_in[11];
    const float* batt = (const float*)d_in[12];
    const float* Wc   = (const float*)d_in[13];
    const float* bc   = (const float*)d_in[14];

    const int E   = in_sizes[2];
    const int N   = in_sizes[4];
    const int Hn  = in_sizes[11];          // 128
    const int FIN = in_sizes[0] / N;       // 128
    const int ED  = in_sizes[6] - 2 * Hn;  // 16
    const int C   = in_sizes[14];          // 10
    const int G   = out_size / C;          // 64

    // Workspace layout (floats)
    float* ws   = (float*)d_ws;
    size_t NH   = (size_t)N * Hn;
    float* zbuf = ws;                 // N*Hn
    float* hA   = zbuf + NH;          // N*Hn
    float* hB   = hA + NH;            // N*Hn
    float* ssrc = hB + NH;            // N
    float* sdst = ssrc + N;           // N
    float* nmax = sdst + N;           // N
    float* nsum = nmax + N;           // N
    float* elog = nsum + N;           // E
    float* hg   = elog + E;           // 3*G*Hn
    float* cnt  = hg + (size_t)3 * G * Hn;  // G

    const int waves_per_block = 8;
    int grid_nodeW = (N + waves_per_block - 1) / waves_per_block;
    int grid_edge  = (E + 255) / 256;
    int grid_edgeW = (E + waves_per_block - 1) / waves_per_block;
    long long nh_tot = (long long)N * Hn;
    int grid_reset = (int)((nh_tot + 255) / 256);
    int grid_fin   = (G * Hn + 255) / 256;

    auto launch_gemm = [&](const float* hin, int Kdim, const float* Wl) {
        dim3 gg((N + 32 * GEMM_TPB - 1) / (32 * GEMM_TPB), Hn / 64);
        size_t shbytes = (size_t)(64 + 32) * (Kdim + 4) * sizeof(float);
        gemm_wmma_f32<<<gg, 256, shbytes, stream>>>(hin, Wl, zbuf, N, Kdim, Hn);
    };

    auto run_layer = [&](const float* hin, int Kdim, const float* Wl,
                         const float* al, float* hnext, float* hgslab) {
        launch_gemm(hin, Kdim, Wl);
        node_scores<<<grid_nodeW, 256, 0, stream>>>(zbuf, al, ssrc, sdst, N, Hn);
        reset_layer<<<grid_reset, 256, 0, stream>>>(hnext, nmax, nsum, hgslab, cnt, N, Hn, G);
        edge_logits<<<grid_edge, 256, 0, stream>>>(ef, al + 2 * Hn, src, dst,
                                                   ssrc, sdst, elog, nmax, E, ED);
        edge_exp<<<grid_edge, 256, 0, stream>>>(dst, elog, nmax, nsum, E);
        aggregate<<<grid_edgeW, 256, 0, stream>>>(zbuf, src, dst, elog, nsum, hnext, E, Hn);
        relu_inplace<<<grid_reset, 256, 0, stream>>>(hnext, nh_tot);
        readout_accum<<<grid_nodeW, 256, 0, stream>>>(hnext, gid, watt, batt,
                                                      hgslab, cnt, N, Hn);
        readout_finalize<<<grid_fin, 256, 0, stream>>>(hgslab, cnt, G, Hn);
    };

    run_layer(x,  FIN, W1, a1, hA, hg + 0 * (size_t)G * Hn);
    run_layer(hA, Hn,  W2, a2, hB, hg + 1 * (size_t)G * Hn);
    run_layer(hB, Hn,  W3, a3, hA, hg + 2 * (size_t)G * Hn);   // reuse hA slab

    classifier<<<G, 32, 0, stream>>>(hg, Wc, bc, (float*)d_out, G, Hn, C);
}